// SoftmaxTreeWithLoss_58987080843957
// MI455X (gfx1250) — compile-verified
//
#include <hip/hip_runtime.h>
#include <hip/hip_bf16.h>
#include <stdint.h>

// Problem constants (fixed by the reference setup)
#define B_DIM     16
#define C_DIM     9200
#define HW_DIM    169            // 13*13
#define N_ROOT    400
#define CHUNK_ROWS  25           // channels per TDM chunk
#define CHUNK_ELEMS (CHUNK_ROWS * HW_DIM)   // 4225 f32 = 16.9 KB
#define N_CHUNKS    (N_ROOT / CHUNK_ROWS)   // 16
#define INV_COUNT  (1.0f / 2704.0f)

#if defined(__gfx1250__) && __has_builtin(__builtin_amdgcn_tensor_load_to_lds) && __has_builtin(__builtin_amdgcn_s_wait_tensorcnt)
#define USE_TDM 1
#else
#define USE_TDM 0
#endif

typedef __attribute__((ext_vector_type(4))) unsigned int uint32x4_t_;
typedef __attribute__((ext_vector_type(8))) int          int32x8_t_;
typedef __attribute__((ext_vector_type(4))) int          int32x4_t_;

#if USE_TDM
// 1-D TDM tile: copy nelem f32 (contiguous) from global -> LDS.
// D# group0: count=1, lds_addr, 57b global addr, type=2.
// D# group1: data_size=2(4B), tensor_dim0=nelem, tensor_dim1=1,
//            tile_dim0=nelem (<=65535), tile_dim1/2=0, dim0_stride=nelem.
__device__ __forceinline__ void tdm_load_1d_f32(unsigned lds_byte_addr,
                                                const float* gsrc,
                                                unsigned nelem) {
  unsigned long long ga = (unsigned long long)(uintptr_t)gsrc;
  uint32x4_t_ g0;
  g0.x = 1u;                                            // count=1, user desc
  g0.y = lds_byte_addr;                                 // bits 63:32
  g0.z = (unsigned)(ga & 0xFFFFFFFFull);                // global_addr lo
  g0.w = (unsigned)((ga >> 32) & 0x01FFFFFFull)         // global_addr hi (57b)
       | (2u << 30);                                    // type = 2 ("image")
  int32x8_t_ g1;
  g1[0] = (int)(2u << 16);                              // data_size=2 -> 4B; mask=0
  g1[1] = (int)((nelem & 0xFFFFu) << 16);               // tensor_dim0[15:0]
  g1[2] = (int)(((nelem >> 16) & 0xFFFFu) | (1u << 16));// tensor_dim0 hi | tensor_dim1=1
  g1[3] = (int)((nelem & 0xFFFFu) << 16);               // tile_dim0 = nelem
  g1[4] = 0;                                            // tile_dim1=0 (1-D), tile_dim2=0
  g1[5] = (int)nelem;                                   // tensor_dim0_stride lo
  g1[6] = 0;                                            // stride hi, dim1_stride lo
  g1[7] = 0;
  int32x4_t_ z4 = {0, 0, 0, 0};
  int32x8_t_ z8 = {0, 0, 0, 0, 0, 0, 0, 0};
  // 6-arg variant on this toolchain: (g0, g1, g2, g3, g4, cpol)
  __builtin_amdgcn_tensor_load_to_lds(g0, g1, z4, z4, z8, 0);
}
#endif

__global__ __launch_bounds__(256) void tree_loss_main(
    const float* __restrict__ x,     // [16, 9200, 13, 13]
    const int*   __restrict__ label, // [16, 13, 13]
    const int*   __restrict__ goff,  // [401]
    const int*   __restrict__ gsz,   // [401]
    const int*   __restrict__ cid,   // [9200]
    const int*   __restrict__ par,   // [9200]
    float*       __restrict__ ws)    // [16] block partial sums
{
  __shared__ float tiles[2][CHUNK_ELEMS];
  __shared__ float red[256];

  const int tid = (int)threadIdx.x;
  const int b   = (int)blockIdx.x;
  const float* xb = x + (size_t)b * C_DIM * HW_DIM;

  // ---- Phase 1: group-0 (root group, 400 channels) online logsumexp ----
  // thread tid (< 169) owns spatial position hw = tid; loads are coalesced
  // across lanes (consecutive hw are contiguous in memory for a channel).
  float m = -3.402823466e38f;
  float s = 0.0f;

#if USE_TDM
  // double-buffered TDM pipeline: DMA chunk c+1 overlaps LSE of chunk c
  if (tid < 32) {
    tdm_load_1d_f32((unsigned)(uintptr_t)&tiles[0][0], xb, CHUNK_ELEMS);
  }
  for (int c = 0; c < N_CHUNKS; ++c) {
    if (tid < 32) __builtin_amdgcn_s_wait_tensorcnt(0);   // chunk c landed
    __syncthreads();                                      // publish to all waves
    if (tid < 32 && (c + 1) < N_CHUNKS) {
      tdm_load_1d_f32((unsigned)(uintptr_t)&tiles[(c + 1) & 1][0],
                      xb + (size_t)(c + 1) * CHUNK_ELEMS, CHUNK_ELEMS);
    }
    if (tid < HW_DIM) {
      const float* t = tiles[c & 1];
#pragma unroll
      for (int r = 0; r < CHUNK_ROWS; ++r) {
        float v  = t[r * HW_DIM + tid];                   // bank-conflict-free
        float mn = fmaxf(m, v);
        s = s * __expf(m - mn) + __expf(v - mn);
        m = mn;
      }
    }
    __syncthreads();   // compute(c) done before buffer (c&1) is overwritten
  }
#else
  for (int c = 0; c < N_CHUNKS; ++c) {
    for (int i = tid; i < CHUNK_ELEMS; i += 256)          // coalesced copy
      tiles[0][i] = xb[(size_t)c * CHUNK_ELEMS + i];
    __syncthreads();
    if (tid < HW_DIM) {
      const float* t = tiles[0];
#pragma unroll
      for (int r = 0; r < CHUNK_ROWS; ++r) {
        float v  = t[r * HW_DIM + tid];
        float mn = fmaxf(m, v);
        s = s * __expf(m - mn) + __expf(v - mn);
        m = mn;
      }
    }
    __syncthreads();
  }
#endif

  // ---- Phase 2: per-position path loss ----
  float lossv = 0.0f;
  if (tid < HW_DIM) {
    const float lse0 = m + __logf(s);
    const int n = label[b * HW_DIM + tid];
    int r;
    if (n >= N_ROOT) {
      // child term: logsumexp over its (contiguous) sibling group
      const int g   = cid[n];
      const int off = goff[g];
      const int sz  = gsz[g];
      float cm = -3.402823466e38f, cs = 0.0f;
      for (int j = 0; j < sz; ++j) {
        float v  = xb[(size_t)(off + j) * HW_DIM + tid];  // coalesced across lanes
        float mn = fmaxf(cm, v);
        cs = cs * __expf(cm - mn) + __expf(v - mn);
        cm = mn;
      }
      lossv = (cm + __logf(cs)) - xb[(size_t)n * HW_DIM + tid];
      r = par[n];
    } else {
      r = n;
    }
    // root term (always present; roots live in group 0)
    lossv += lse0 - xb[(size_t)r * HW_DIM + tid];
  }

  // ---- deterministic block reduction ----
  red[tid] = lossv;
  __syncthreads();
  for (int off = 128; off > 0; off >>= 1) {
    if (tid < off) red[tid] += red[tid + off];
    __syncthreads();
  }
  if (tid == 0) ws[b] = red[0];
}

__global__ void tree_loss_finish(const float* __restrict__ ws,
                                 float* __restrict__ out) {
  if (threadIdx.x == 0) {
    float t = 0.0f;
#pragma unroll
    for (int i = 0; i < B_DIM; ++i) t += ws[i];
    out[0] = t * INV_COUNT;   // LOSS_WEIGHT == 1.0
  }
}

extern "C" void kernel_launch(void* const* d_in, const int* in_sizes, int n_in,
                              void* d_out, int out_size, void* d_ws, size_t ws_size,
                              hipStream_t stream) {
  (void)in_sizes; (void)n_in; (void)out_size; (void)ws_size;
  const float* x     = (const float*)d_in[0];
  const int*   label = (const int*)d_in[1];
  const int*   goff  = (const int*)d_in[2];
  const int*   gsz   = (const int*)d_in[3];
  const int*   cid   = (const int*)d_in[4];
  const int*   par   = (const int*)d_in[5];
  float*       ws    = (float*)d_ws;

  tree_loss_main<<<B_DIM, 256, 0, stream>>>(x, label, goff, gsz, cid, par, ws);
  tree_loss_finish<<<1, 32, 0, stream>>>(ws, (float*)d_out);
}